// Detection_15891378995707
// MI455X (gfx1250) — compile-verified
//
#include <hip/hip_runtime.h>
#include <hip/hip_bf16.h>
#include <stdint.h>

#define BATCH 8
#define NP 24564
#define NC 81
#define TOPK 200
#define CAND 512
#define HBINS 2048
#define CONF_T 0.01f
#define NMS_T 0.45f
#define TPRI 32

typedef unsigned int u32x4 __attribute__((ext_vector_type(4)));
typedef int          i32x8 __attribute__((ext_vector_type(8)));
typedef int          i32x4 __attribute__((ext_vector_type(4)));

// ---------------------------------------------------------------------------
// Kernel 1: decode prior boxes with loc deltas.
// boxes[b][p] = decode(loc[b][p], prior_center_size[p])   (float4 per box)
// ---------------------------------------------------------------------------
__global__ __launch_bounds__(256) void decode_boxes_k(
    const float* __restrict__ loc, const float* __restrict__ prior,
    float* __restrict__ boxes)
{
    int idx = blockIdx.x * blockDim.x + threadIdx.x;
    if (idx >= BATCH * NP) return;
    int b = idx / NP;
    int p = idx - b * NP;

    float4 l  = ((const float4*)loc)[(size_t)b * NP + p];
    float4 pr = ((const float4*)prior)[p];           // corners row of prior[0,0]

    float cx = 0.5f * (pr.x + pr.z);
    float cy = 0.5f * (pr.y + pr.w);
    float w  = pr.z - pr.x;
    float h  = pr.w - pr.y;

    float x  = cx + l.x * 0.1f * w;
    float y  = cy + l.y * 0.1f * h;
    float ww = w * __expf(l.z * 0.2f);
    float hh = h * __expf(l.w * 0.2f);

    float4 o;
    o.x = x - 0.5f * ww;
    o.y = y - 0.5f * hh;
    o.z = o.x + ww;       // x2y2 = wh + x1y1 (matches reference ordering)
    o.w = o.y + hh;
    ((float4*)boxes)[(size_t)b * NP + p] = o;
}

// ---------------------------------------------------------------------------
// Kernel 2: transpose conf [B][NP][81] -> conf_t [B][81][NP] via LDS tile so
// each class row becomes a contiguous 98KB stream (TDM-friendly, coalesced).
// ---------------------------------------------------------------------------
__global__ __launch_bounds__(256) void transpose_conf_k(
    const float* __restrict__ conf, float* __restrict__ conf_t)
{
    __shared__ float tile[TPRI * NC];       // 32 priors x 81 classes = 2592
    int b  = blockIdx.y;
    int p0 = blockIdx.x * TPRI;
    const float* src = conf + (size_t)b * NP * NC + (size_t)p0 * NC;
    int nelem = (NP - p0 < TPRI ? NP - p0 : TPRI) * NC;

    for (int e = threadIdx.x; e < TPRI * NC; e += blockDim.x)
        tile[e] = (e < nelem) ? src[e] : 0.0f;

    if (threadIdx.x == 0 && p0 + TPRI < NP)
        __builtin_prefetch(src + TPRI * NC, 0, 1);   // global_prefetch_b8

    __syncthreads();

    for (int e = threadIdx.x; e < TPRI * NC; e += blockDim.x) {
        int c  = e >> 5;        // 2592 = 81 * 32
        int tp = e & 31;
        if (p0 + tp < NP)
            conf_t[((size_t)b * NC + c) * NP + p0 + tp] = tile[tp * NC + c];
    }
}

// ---------------------------------------------------------------------------
// Kernel 3: per (batch, class) workgroup (640 WGs x 256 threads):
//   TDM-load 24564-float class row into LDS (overlapped with histogram init),
//   two-level radix-select pivot, gather candidates, 512-wide bitonic sort,
//   sequential NMS, write rows.
// ---------------------------------------------------------------------------
__global__ __launch_bounds__(256) void select_nms_k(
    const float* __restrict__ conf_t, const float* __restrict__ boxes,
    float* __restrict__ out)
{
    extern __shared__ char smem_raw[];
    float*  s_scores = (float*)smem_raw;               // NP floats   (98256 B)
    int*    s_hist   = (int*)(s_scores + NP);          // 2048        (8192 B)
    float*  s_cs     = (float*)(s_hist + HBINS);       // 512 scores
    int*    s_ci     = (int*)(s_cs + CAND);            // 512 indices
    float4* s_box    = (float4*)(s_ci + CAND);         // 200 boxes (16B aligned)
    float*  s_area   = (float*)(s_box + TOPK);         // 200
    int*    s_keep   = (int*)(s_area + TOPK);          // 200
    int*    s_ctrl   = s_keep + TOPK;                  // [0]=t1 [1]=above [2]=t2 [3]=cnt [4]=sup

    const int tid = threadIdx.x;
    const int bc  = blockIdx.x;          // 0 .. BATCH*80-1
    const int b   = bc / 80;
    const int c   = bc - b * 80 + 1;     // class 1..80 (skip background)
    const float* srow = conf_t + ((size_t)b * NC + c) * NP;

    // ---- Stage class row into LDS via TDM; overlap DMA with hist init ---
#if defined(__gfx1250__) && __has_builtin(__builtin_amdgcn_tensor_load_to_lds)
    if ((tid >> 5) == 0) {               // wave 0 issues the TDM descriptor
        unsigned lds_addr = (unsigned)(uintptr_t)(void*)s_scores;   // addr[31:0] = LDS offset
        unsigned long long ga = (unsigned long long)(uintptr_t)srow;

        // D# group 0: count=1 | lds_addr | global_addr[56:0] | type=2
        u32x4 g0 = { 1u,
                     lds_addr,
                     (unsigned)(ga & 0xFFFFFFFFu),
                     (unsigned)((ga >> 32) & 0x01FFFFFFu) | (2u << 30) };

        // D# group 1: data_size=4B, tensor_dim0=NP, tensor_dim1=1,
        //             tile_dim0=NP, tile_dim1=1, stride0=NP
        i32x8 g1 = { (int)(2u << 16),                          // data_size=2 (4B)
                     (int)(((unsigned)NP & 0xFFFFu) << 16),    // dim0[15:0]
                     (int)(((unsigned)NP >> 16) | (1u << 16)), // dim0[31:16] | dim1 lo
                     (int)(((unsigned)NP & 0xFFFFu) << 16),    // dim1 hi=0 | tile_dim0
                     1,                                        // tile_dim1=1, tile_dim2=0
                     NP,                                       // stride0[31:0]
                     0, 0 };
        i32x4 g2 = { 0, 0, 0, 0 };
        i32x4 g3 = { 0, 0, 0, 0 };
        i32x8 g4 = { 0, 0, 0, 0, 0, 0, 0, 0 };
        __builtin_amdgcn_tensor_load_to_lds(g0, g1, g2, g3, g4, 0);
    }
    // All 8 waves clear the histogram + control state while the DMA runs.
    for (int i = tid; i < HBINS; i += 256) s_hist[i] = 0;
    if (tid == 0) s_ctrl[3] = 0;
    if ((tid >> 5) == 0)
        __builtin_amdgcn_s_wait_tensorcnt(0);   // issuing wave drains TENSORcnt
    __syncthreads();                            // data + zeroed hist visible to all
#else
    {
        const float4* s4 = (const float4*)srow;     // NP % 4 == 0
        float4* d4 = (float4*)s_scores;
        for (int i = tid; i < NP / 4; i += 256) d4[i] = s4[i];
        for (int i = tid; i < HBINS; i += 256) s_hist[i] = 0;
        if (tid == 0) s_ctrl[3] = 0;
        __syncthreads();
    }
#endif

    // ---- Level-1 radix histogram on key[31:21] --------------------------
    for (int i = tid; i < NP; i += 256) {
        float s = s_scores[i];
        if (s > CONF_T) atomicAdd(&s_hist[__float_as_uint(s) >> 21], 1);
    }
    __syncthreads();
    if (tid == 0) {
        int acc = 0, t1 = 0;
        for (int bin = HBINS - 1; bin >= 0; --bin) {
            int h = s_hist[bin];
            if (acc + h >= TOPK || bin == 0) { t1 = bin; break; }
            acc += h;
        }
        s_ctrl[0] = t1; s_ctrl[1] = acc;
    }
    __syncthreads();
    const int t1 = s_ctrl[0];
    const int cntAbove = s_ctrl[1];

    // ---- Level-2 histogram on key[20:10] within bin t1 ------------------
    for (int i = tid; i < HBINS; i += 256) s_hist[i] = 0;
    __syncthreads();
    for (int i = tid; i < NP; i += 256) {
        float s = s_scores[i];
        unsigned k = __float_as_uint(s);
        if (s > CONF_T && (int)(k >> 21) == t1)
            atomicAdd(&s_hist[(k >> 10) & (HBINS - 1)], 1);
    }
    __syncthreads();
    if (tid == 0) {
        int rem = TOPK - cntAbove;
        int acc = 0, t2 = 0;
        for (int bin = HBINS - 1; bin >= 0; --bin) {
            int h = s_hist[bin];
            if (acc + h >= rem || bin == 0) { t2 = bin; break; }
            acc += h;
        }
        s_ctrl[2] = t2;
    }
    __syncthreads();
    const unsigned pivot = ((unsigned)t1 << 21) | ((unsigned)s_ctrl[2] << 10);

    // ---- Gather candidates >= pivot ------------------------------------
    for (int i = tid; i < NP; i += 256) {
        float s = s_scores[i];
        if (s > CONF_T && __float_as_uint(s) >= pivot) {
            int slot = atomicAdd(&s_ctrl[3], 1);
            if (slot < CAND) { s_cs[slot] = s; s_ci[slot] = i; }
        }
    }
    __syncthreads();
    int ncand = s_ctrl[3] < CAND ? s_ctrl[3] : CAND;
    for (int i = tid; i < CAND; i += 256)
        if (i >= ncand) { s_cs[i] = -__builtin_inff(); s_ci[i] = 0x7FFFFFFF; }

    // ---- Bitonic sort, descending by score (tie: lower index) ----------
    for (int k = 2; k <= CAND; k <<= 1) {
        for (int j = k >> 1; j > 0; j >>= 1) {
            __syncthreads();
            for (int i = tid; i < CAND; i += 256) {
                int ix = i ^ j;
                if (ix > i) {
                    float si = s_cs[i], sx = s_cs[ix];
                    int   ii = s_ci[i], ixi = s_ci[ix];
                    bool iBetter = (si > sx) || (si == sx && ii < ixi);
                    bool wantBetterAtI = ((i & k) == 0);
                    if (iBetter != wantBetterAtI) {
                        s_cs[i] = sx; s_cs[ix] = si;
                        s_ci[i] = ixi; s_ci[ix] = ii;
                    }
                }
            }
        }
    }
    __syncthreads();

    // ---- Fetch top-200 boxes, precompute areas -------------------------
    for (int i = tid; i < TOPK; i += 256) {
        float s = s_cs[i];
        float4 bb = make_float4(0.f, 0.f, 0.f, 0.f);
        if (s > CONF_T)
            bb = ((const float4*)boxes)[(size_t)b * NP + s_ci[i]];
        s_box[i]  = bb;
        s_area[i] = (bb.z - bb.x) * (bb.w - bb.y);
        s_keep[i] = 0;
    }
    __syncthreads();

    // ---- Sequential greedy NMS (parallel over j, OR-reduced in LDS) ----
    for (int i = 0; i < TOPK; ++i) {
        if (tid == 0) s_ctrl[4] = 0;
        __syncthreads();
        int j = tid;
        if (j < i && s_keep[j]) {
            float4 bi = s_box[i], bj = s_box[j];
            float xx1 = fmaxf(bi.x, bj.x), yy1 = fmaxf(bi.y, bj.y);
            float xx2 = fminf(bi.z, bj.z), yy2 = fminf(bi.w, bj.w);
            float inter = fmaxf(xx2 - xx1, 0.f) * fmaxf(yy2 - yy1, 0.f);
            float uni = s_area[i] + s_area[j] - inter;
            float iou = inter / (uni > 0.f ? uni : 1.f);
            if (iou > NMS_T) atomicOr(&s_ctrl[4], 1);
        }
        __syncthreads();
        if (tid == 0)
            s_keep[i] = (s_cs[i] > CONF_T && !s_ctrl[4]) ? 1 : 0;
        __syncthreads();
    }
    // KEEP_TOP_K == TOP_K -> cumsum filter is identity; skip.

    // ---- Emit [img, cls, score, x1, y1, x2, y2] rows --------------------
    float* orow = out + (size_t)bc * TOPK * 7;
    for (int i = tid; i < TOPK; i += 256) {
        float* r = orow + (size_t)i * 7;
        if (s_keep[i]) {
            float4 bb = s_box[i];
            r[0] = (float)b; r[1] = (float)c; r[2] = s_cs[i];
            r[3] = bb.x; r[4] = bb.y; r[5] = bb.z; r[6] = bb.w;
        } else {
            #pragma unroll
            for (int q = 0; q < 7; ++q) r[q] = 0.f;
        }
    }
}

// ---------------------------------------------------------------------------
extern "C" void kernel_launch(void* const* d_in, const int* in_sizes, int n_in,
                              void* d_out, int out_size, void* d_ws, size_t ws_size,
                              hipStream_t stream) {
    const float* loc   = (const float*)d_in[0];   // (B, NP*4)
    const float* conf  = (const float*)d_in[1];   // (B, NP*81)
    const float* prior = (const float*)d_in[2];   // (1, 2, NP*4)
    float* out = (float*)d_out;                   // (B*80*200, 7)

    float* boxes  = (float*)d_ws;                         // B*NP*4 floats
    float* conf_t = boxes + (size_t)BATCH * NP * 4;       // B*81*NP floats

    decode_boxes_k<<<(BATCH * NP + 255) / 256, 256, 0, stream>>>(loc, prior, boxes);

    dim3 tg((NP + TPRI - 1) / TPRI, BATCH);
    transpose_conf_k<<<tg, 256, 0, stream>>>(conf, conf_t);

    size_t shmem = (size_t)NP * 4 + (size_t)HBINS * 4 + (size_t)CAND * 8 +
                   (size_t)TOPK * (16 + 4 + 4) + 64;
    static bool attr_set = false;
    if (!attr_set) {  // host-side one-time attribute; does not affect device work
        (void)hipFuncSetAttribute((const void*)select_nms_k,
                                  hipFuncAttributeMaxDynamicSharedMemorySize, (int)shmem);
        attr_set = true;
    }
    select_nms_k<<<BATCH * 80, 256, shmem, stream>>>(conf_t, boxes, out);
}